// KANLayer_48962627174734
// MI455X (gfx1250) — compile-verified
//
#include <hip/hip_runtime.h>

typedef __attribute__((ext_vector_type(2))) float v2f;
typedef __attribute__((ext_vector_type(8))) float v8f;

#define B_   512
#define I_   256
#define O_   256
#define K_   8
// NORM = 1/(0.5*sqrt(2*pi)); exponent coeff = -1/(2*sigma^2) = -2
#define NORM_ 0.7978845608028654f

constexpr int MT = 16;            // B-rows per workgroup (one WMMA M tile)
constexpr int NT = 64;            // O-cols per workgroup (4 waves x 16)
constexpr int IB = 16;            // input features per chunk
constexpr int KC = IB * (K_ + 1); // 144 reduction columns per chunk (8 basis + 1 silu)

__global__ __launch_bounds__(128)
void kan_wmma_f32_kernel(const float* __restrict__ x,
                         const float* __restrict__ cp,
                         const float* __restrict__ coef,
                         const float* __restrict__ w,
                         float* __restrict__ out)
{
    __shared__ float xs[MT][IB];        // 1 KB
    __shared__ float Abuf[MT][KC];      // 16 x 144 f32 = 9 KB
    __shared__ float Bbuf[KC][NT];      // 144 x 64 f32 = 36 KB

    const int tid  = threadIdx.x;
    const int lane = tid & 31;
    const int wave = tid >> 5;               // 0..3
    const int b0   = blockIdx.x * MT;        // batch-row tile base
    const int oc0  = blockIdx.y * NT;        // output-col tile base
    const int n0   = wave * 16;              // this wave's 16 cols inside NT

    // WMMA f32 16x16x4 fragment addressing (ISA 7.12.2):
    // A (16x4): lanes 0-15 hold K0/K1 for M=lane, lanes 16-31 hold K2/K3 for M=lane-16
    const int rowA  = lane & 15;
    const int kselA = (lane < 16) ? 0 : 2;
    // B (4x16): lanes 0-15 hold rows K0/K1 at N=lane, lanes 16-31 rows K2/K3 at N=lane-16
    const int colB  = lane & 15;
    const int r0B   = (lane < 16) ? 0 : 2;

    v8f acc = {};

    for (int ic0 = 0; ic0 < I_; ic0 += IB) {
        __syncthreads();   // previous iteration's WMMA reads of LDS are done

        // stage x tile [16 b-rows][16 i-cols]
        for (int e = tid; e < MT * IB; e += 128) {
            int bl = e >> 4, il = e & 15;
            xs[bl][il] = x[(b0 + bl) * I_ + (ic0 + il)];
        }
        __syncthreads();

        // A-chunk: basis[b,i,k] for k<8, silu(x[b,i]) for k==8
        for (int e = tid; e < MT * KC; e += 128) {
            int bl = e / KC, c = e - bl * KC;
            int il = c / (K_ + 1), k = c - il * (K_ + 1);
            float xv = xs[bl][il];
            float val;
            if (k < K_) {
                // cp is broadcast over O in the reference; read the o=0 entry
                float d = xv - cp[(ic0 + il) * (O_ * K_) + k];
                val = NORM_ * __expf(-2.0f * d * d);
            } else {
                val = xv * (1.0f / (1.0f + __expf(-xv)));   // silu
            }
            Abuf[bl][c] = val;
        }

        // B-chunk: w[i,o]*coef[i,o,k] for k<8, w[i,o] for the silu slot
        for (int e = tid; e < IB * NT; e += 128) {
            int il = e >> 6, ol = e & 63;
            int i = ic0 + il, o = oc0 + ol;
            float wv = w[i * O_ + o];
            const float4* c4 = (const float4*)(coef + (size_t)(i * O_ + o) * K_);
            float4 ca = c4[0], cb = c4[1];
            int r = il * (K_ + 1);
            Bbuf[r + 0][ol] = wv * ca.x;
            Bbuf[r + 1][ol] = wv * ca.y;
            Bbuf[r + 2][ol] = wv * ca.z;
            Bbuf[r + 3][ol] = wv * ca.w;
            Bbuf[r + 4][ol] = wv * cb.x;
            Bbuf[r + 5][ol] = wv * cb.y;
            Bbuf[r + 6][ol] = wv * cb.z;
            Bbuf[r + 7][ol] = wv * cb.w;
            Bbuf[r + 8][ol] = wv;
        }
        __syncthreads();

        // 36 WMMA steps over the 144-wide reduction chunk
        #pragma unroll
        for (int kk = 0; kk < KC; kk += 4) {
            v2f a, b;
            a.x = Abuf[rowA][kk + kselA];
            a.y = Abuf[rowA][kk + kselA + 1];
            b.x = Bbuf[kk + r0B][n0 + colB];
            b.y = Bbuf[kk + r0B + 1][n0 + colB];
            acc = __builtin_amdgcn_wmma_f32_16x16x4_f32(
                /*neg_a=*/false, a, /*neg_b=*/false, b,
                /*c_mod=*/(short)0, acc, /*reuse_a=*/false, /*reuse_b=*/false);
        }
    }

    // C/D layout: VGPR r -> M = r (lanes 0-15) or r+8 (lanes 16-31), N = lane&15
    const int ocol = oc0 + n0 + colB;
    const int brow = b0 + ((lane < 16) ? 0 : 8);
    #pragma unroll
    for (int r = 0; r < 8; ++r) {
        out[(brow + r) * O_ + ocol] = acc[r];
    }
}

extern "C" void kernel_launch(void* const* d_in, const int* in_sizes, int n_in,
                              void* d_out, int out_size, void* d_ws, size_t ws_size,
                              hipStream_t stream) {
    (void)in_sizes; (void)n_in; (void)d_ws; (void)ws_size; (void)out_size;
    const float* x    = (const float*)d_in[0];   // [B,I]
    const float* cp   = (const float*)d_in[1];   // [I,O,K]
    const float* coef = (const float*)d_in[2];   // [I,O,K]
    const float* w    = (const float*)d_in[3];   // [I,O]
    float* out = (float*)d_out;                  // [B,O]

    dim3 grid(B_ / MT, O_ / NT);                 // (32, 4)
    kan_wmma_f32_kernel<<<grid, 128, 0, stream>>>(x, cp, coef, w, out);
}